// BoundaryLoss_85761906966839
// MI455X (gfx1250) — compile-verified
//
#include <hip/hip_runtime.h>

// BoundaryLoss reduction for MI455X (gfx1250, wave32).
// HBM-bound: ~100.7 MB streamed once -> ~4.3 us floor at 23.3 TB/s.
// Strategy: 128-bit NT loads with deep MLP (counted, unrolled loop -> many
// outstanding global_load_b128 per wave), fp32 FMA into 4 rotating
// accumulators, wave reduction via V_WMMA_F32_16X16X4_F32 (ones-matrix
// trick), two-stage deterministic output (no atomics, no ws reuse hazards).

typedef __attribute__((ext_vector_type(2))) float v2f;
typedef __attribute__((ext_vector_type(4))) float v4f;
typedef __attribute__((ext_vector_type(8))) float v8f;

#define PLANE_F4   65536                    // 512*512/4 float4 per (n,c) plane
#define NC_DENSE   48                       // 16 batches * 3 foreground classes
#define TOTAL_F4   (NC_DENSE * PLANE_F4)    // 3,145,728 float4 pairs
#define BLOCKS     1024
#define TPB        256
#define ITERS      (TOTAL_F4 / (BLOCKS * TPB))   // exactly 12
#define STRIDE_PLANES 4                     // grid stride = 4 dense planes

// Sum of all 32 lane values, result broadcast to every lane.
// A (16x4 f32): VGPR0 = acc, VGPR1 = 0.  B = all ones (layout-independent).
// D[m][n] = acc_m + acc_{m+16}; lane l<16 holds rows 0..7 of col l, lane
// l+16 holds rows 8..15 -> per-lane 8-add + shfl_xor(16) = full 32-lane sum.
__device__ __forceinline__ float wave_reduce_f32(float acc) {
  v2f a; a[0] = acc;  a[1] = 0.0f;
  v2f b; b[0] = 1.0f; b[1] = 1.0f;
  v8f c = {};
  c = __builtin_amdgcn_wmma_f32_16x16x4_f32(
      /*neg_a=*/false, a, /*neg_b=*/false, b,
      /*c_mod=*/(short)0, c, /*reuse_a=*/false, /*reuse_b=*/false);
  float s = ((c[0] + c[1]) + (c[2] + c[3])) + ((c[4] + c[5]) + (c[6] + c[7]));
  s += __shfl_xor(s, 16, 32);
  return s;
}

// Dense plane d in [0,48) -> memory float4 index of its first element.
// memory plane m = 4n + c = d + d/3 + 1  (skips background class 0).
__device__ __forceinline__ int plane_to_mem(int d) {
  return (d + d / 3 + 1) << 16;
}

__global__ void __launch_bounds__(TPB)
boundary_partial_kernel(const v4f* __restrict__ sm,
                        const v4f* __restrict__ dm,
                        float* __restrict__ partial) {
  const int tid    = blockIdx.x * TPB + (int)threadIdx.x;   // 0 .. 262143
  const int within = tid & (PLANE_F4 - 1);
  const int plane0 = tid >> 16;                             // 0..3

  float acc0 = 0.0f, acc1 = 0.0f, acc2 = 0.0f, acc3 = 0.0f;
#pragma unroll
  for (int it = 0; it < ITERS; it += 4) {
    const int m0 = plane_to_mem(plane0 + (it + 0) * STRIDE_PLANES) | within;
    const int m1 = plane_to_mem(plane0 + (it + 1) * STRIDE_PLANES) | within;
    const int m2 = plane_to_mem(plane0 + (it + 2) * STRIDE_PLANES) | within;
    const int m3 = plane_to_mem(plane0 + (it + 3) * STRIDE_PLANES) | within;
    const v4f a0 = __builtin_nontemporal_load(&sm[m0]);   // global_load_b128 NT
    const v4f b0 = __builtin_nontemporal_load(&dm[m0]);
    const v4f a1 = __builtin_nontemporal_load(&sm[m1]);
    const v4f b1 = __builtin_nontemporal_load(&dm[m1]);
    const v4f a2 = __builtin_nontemporal_load(&sm[m2]);
    const v4f b2 = __builtin_nontemporal_load(&dm[m2]);
    const v4f a3 = __builtin_nontemporal_load(&sm[m3]);
    const v4f b3 = __builtin_nontemporal_load(&dm[m3]);
    acc0 = fmaf(a0[0], b0[0], acc0); acc0 = fmaf(a0[1], b0[1], acc0);
    acc0 = fmaf(a0[2], b0[2], acc0); acc0 = fmaf(a0[3], b0[3], acc0);
    acc1 = fmaf(a1[0], b1[0], acc1); acc1 = fmaf(a1[1], b1[1], acc1);
    acc1 = fmaf(a1[2], b1[2], acc1); acc1 = fmaf(a1[3], b1[3], acc1);
    acc2 = fmaf(a2[0], b2[0], acc2); acc2 = fmaf(a2[1], b2[1], acc2);
    acc2 = fmaf(a2[2], b2[2], acc2); acc2 = fmaf(a2[3], b2[3], acc2);
    acc3 = fmaf(a3[0], b3[0], acc3); acc3 = fmaf(a3[1], b3[1], acc3);
    acc3 = fmaf(a3[2], b3[2], acc3); acc3 = fmaf(a3[3], b3[3], acc3);
  }
  const float acc = (acc0 + acc1) + (acc2 + acc3);

  // No divergence above: EXEC is all-1s here (WMMA requirement).
  const float wsum = wave_reduce_f32(acc);

  __shared__ float lds[TPB / 32];
  const int lane = threadIdx.x & 31;
  const int wave = threadIdx.x >> 5;
  if (lane == 0) lds[wave] = wsum;
  __syncthreads();
  if (threadIdx.x == 0) {
    float s = lds[0];
#pragma unroll
    for (int i = 1; i < TPB / 32; ++i) s += lds[i];
    partial[blockIdx.x] = s;
  }
}

// One full wave32: EXEC all-1s throughout, 1024/32 = 32 uniform loads/lane.
__global__ void __launch_bounds__(32)
boundary_final_kernel(const float* __restrict__ partial,
                      float* __restrict__ out) {
  float acc = 0.0f;
#pragma unroll
  for (int i = (int)threadIdx.x; i < BLOCKS; i += 32) acc += partial[i];
  const float total = wave_reduce_f32(acc);
  if (threadIdx.x == 0) out[0] = total * (1.0f / 48.0f);
}

extern "C" void kernel_launch(void* const* d_in, const int* in_sizes, int n_in,
                              void* d_out, int out_size, void* d_ws, size_t ws_size,
                              hipStream_t stream) {
  const v4f* sm = (const v4f*)d_in[0];       // softmax_output, f32 NCHW
  // d_in[1] = target (int32) is unused on the precomputed-distance-map path.
  const v4f* dm = (const v4f*)d_in[2];       // distance_maps, f32 NCHW
  float* partial = (float*)d_ws;             // 1024 floats = 4 KB scratch
  float* out     = (float*)d_out;

  boundary_partial_kernel<<<BLOCKS, TPB, 0, stream>>>(sm, dm, partial);
  boundary_final_kernel<<<1, 32, 0, stream>>>(partial, out);
}